// GroupCategorical_48361331753647
// MI455X (gfx1250) — compile-verified
//
#include <hip/hip_runtime.h>
#include <hip/hip_bf16.h>

// ---------------------------------------------------------------------------
// Grouped log-softmax (segment_max + segment_sum(exp) + normalize), sorted idx
// Memory-bound: ~450MB traffic -> ~20us at 23.3 TB/s on MI455X.
// CDNA5 paths: GLOBAL_LOAD_ASYNC_TO_LDS_B128 staging + s_wait_asynccnt,
// wave32 blocks, nontemporal streaming store on the final pass.
// ---------------------------------------------------------------------------

#define BLOCK 256
#define RUN   16
#define TILE  (BLOCK * RUN)   // 4096 elements per block

typedef float __attribute__((ext_vector_type(4))) f32x4;
typedef int   __attribute__((ext_vector_type(4))) i32x4;

// ----- async global->LDS support (guarded; falls back to plain b128 loads) --
#if defined(__has_builtin)
#if __has_builtin(__builtin_amdgcn_global_load_async_to_lds_b128)
#define ASYNC_LDS 1
#endif
#endif
#ifndef ASYNC_LDS
#define ASYNC_LDS 0
#endif

#if ASYNC_LDS
// builtin prototype (from hipcc diagnostic): param0 = i32x4 addrspace(1)*,
// param1 = i32x4 addrspace(3)*, then imm offset, imm cpol.
typedef __attribute__((address_space(1))) i32x4 gv4i_t;
typedef __attribute__((address_space(3))) i32x4 lv4i_t;
__device__ __forceinline__ void async_copy16(void* lds, const void* g) {
  __builtin_amdgcn_global_load_async_to_lds_b128((gv4i_t*)(void*)g,
                                                 (lv4i_t*)lds, 0, 0);
}
#endif

__device__ __forceinline__ void wait_async_all() {
#if defined(__has_builtin) && __has_builtin(__builtin_amdgcn_s_wait_asynccnt)
  __builtin_amdgcn_s_wait_asynccnt(0);
#else
  asm volatile("s_wait_asynccnt 0" ::: "memory");
#endif
}

// ----- order-preserving u32 encoding of float (for native u32 atomic max) ---
__device__ __forceinline__ unsigned fenc(float f) {
  unsigned u = __float_as_uint(f);
  return (u & 0x80000000u) ? ~u : (u | 0x80000000u);
}
__device__ __forceinline__ float fdec(unsigned u) {
  u = (u & 0x80000000u) ? (u & 0x7FFFFFFFu) : ~u;
  return __uint_as_float(u);
}
#define FENC_NEG_INF 0x007FFFFFu  // fenc(-inf)

__device__ __forceinline__ void atomicAddF(float* p, float v) {
  __hip_atomic_fetch_add(p, v, __ATOMIC_RELAXED, __HIP_MEMORY_SCOPE_AGENT);
}

// ---------------------------------------------------------------------------
// K0: init per-segment accumulators (S read from device scalar)
// ---------------------------------------------------------------------------
__global__ void k_init(unsigned* __restrict__ maxEnc, float* __restrict__ sumExp,
                       const int* __restrict__ nseg) {
  const int S = *nseg;
  for (int i = blockIdx.x * blockDim.x + threadIdx.x; i < S;
       i += gridDim.x * blockDim.x) {
    maxEnc[i] = FENC_NEG_INF;
    sumExp[i] = 0.0f;
  }
}

// ---------------------------------------------------------------------------
// K1: segment max. LDS-staged tile, run-length-compressed u32-max atomics.
// ---------------------------------------------------------------------------
__global__ __launch_bounds__(BLOCK) void k_segmax(
    const float* __restrict__ logits, const int* __restrict__ index,
    unsigned* __restrict__ maxEnc, int n) {
  __shared__ float smL[TILE];
  __shared__ int   smI[TILE];
  const int base = blockIdx.x * TILE;
  const int t    = threadIdx.x;

  if (base + TILE <= n) {
#if ASYNC_LDS
#pragma unroll
    for (int k = 0; k < TILE / BLOCK / 4; ++k) {
      const int off = (t + k * BLOCK) * 4;
      async_copy16(&smL[off], &logits[base + off]);
      async_copy16(&smI[off], &index[base + off]);
    }
    wait_async_all();
    __syncthreads();
#else
#pragma unroll
    for (int k = 0; k < TILE / BLOCK / 4; ++k) {
      const int off = (t + k * BLOCK) * 4;
      *(f32x4*)&smL[off] = *(const f32x4*)&logits[base + off];
      *(i32x4*)&smI[off] = *(const i32x4*)&index[base + off];
    }
    __syncthreads();
#endif
    const int s0 = t * RUN;
    int   cur = smI[s0];
    float m   = smL[s0];
#pragma unroll
    for (int k = 1; k < RUN; ++k) {
      const int   ix = smI[s0 + k];
      const float v  = smL[s0 + k];
      if (ix == cur) {
        m = fmaxf(m, v);
      } else {
        atomicMax(&maxEnc[cur], fenc(m));
        cur = ix;
        m   = v;
      }
    }
    atomicMax(&maxEnc[cur], fenc(m));
  } else {
    const int lim = (base + TILE < n) ? (base + TILE) : n;
    for (int i = base + t; i < lim; i += BLOCK)
      atomicMax(&maxEnc[index[i]], fenc(logits[i]));
  }
}

// ---------------------------------------------------------------------------
// K2: segment sum of exp(logit - max). One max gather per run (L2 resident).
// ---------------------------------------------------------------------------
__global__ __launch_bounds__(BLOCK) void k_sumexp(
    const float* __restrict__ logits, const int* __restrict__ index,
    const unsigned* __restrict__ maxEnc, float* __restrict__ sumExp, int n) {
  __shared__ float smL[TILE];
  __shared__ int   smI[TILE];
  const int base = blockIdx.x * TILE;
  const int t    = threadIdx.x;

  if (base + TILE <= n) {
#if ASYNC_LDS
#pragma unroll
    for (int k = 0; k < TILE / BLOCK / 4; ++k) {
      const int off = (t + k * BLOCK) * 4;
      async_copy16(&smL[off], &logits[base + off]);
      async_copy16(&smI[off], &index[base + off]);
    }
    wait_async_all();
    __syncthreads();
#else
#pragma unroll
    for (int k = 0; k < TILE / BLOCK / 4; ++k) {
      const int off = (t + k * BLOCK) * 4;
      *(f32x4*)&smL[off] = *(const f32x4*)&logits[base + off];
      *(i32x4*)&smI[off] = *(const i32x4*)&index[base + off];
    }
    __syncthreads();
#endif
    const int s0  = t * RUN;
    int   cur = smI[s0];
    float mv  = fdec(maxEnc[cur]);
    float acc = __expf(smL[s0] - mv);
#pragma unroll
    for (int k = 1; k < RUN; ++k) {
      const int   ix = smI[s0 + k];
      const float v  = smL[s0 + k];
      if (ix == cur) {
        acc += __expf(v - mv);
      } else {
        atomicAddF(&sumExp[cur], acc);
        cur = ix;
        mv  = fdec(maxEnc[cur]);
        acc = __expf(v - mv);
      }
    }
    atomicAddF(&sumExp[cur], acc);
  } else {
    const int lim = (base + TILE < n) ? (base + TILE) : n;
    for (int i = base + t; i < lim; i += BLOCK) {
      const int ix = index[i];
      atomicAddF(&sumExp[ix], __expf(logits[i] - fdec(maxEnc[ix])));
    }
  }
}

// ---------------------------------------------------------------------------
// K3: per-segment finalize c[s] = max[s] + log(sum[s])  (in place in sumExp)
// ---------------------------------------------------------------------------
__global__ void k_segfinal(const unsigned* __restrict__ maxEnc,
                           float* __restrict__ sumExp,
                           const int* __restrict__ nseg) {
  const int S = *nseg;
  for (int i = blockIdx.x * blockDim.x + threadIdx.x; i < S;
       i += gridDim.x * blockDim.x) {
    sumExp[i] = fdec(maxEnc[i]) + __logf(sumExp[i]);
  }
}

// ---------------------------------------------------------------------------
// K4: streaming normalize  out[i] = logits[i] - c[index[i]]
// b128 loads, nontemporal b128 store (streaming, never reused)
// ---------------------------------------------------------------------------
__global__ __launch_bounds__(BLOCK) void k_write(
    const float* __restrict__ logits, const int* __restrict__ index,
    const float* __restrict__ c, float* __restrict__ out, int n) {
  const int i4 = (blockIdx.x * BLOCK + threadIdx.x) * 4;
  if (i4 + 4 <= n) {
    const f32x4 l  = *(const f32x4*)&logits[i4];
    const i32x4 ix = *(const i32x4*)&index[i4];
    f32x4 o;
    o.x = l.x - c[ix.x];
    o.y = l.y - c[ix.y];
    o.z = l.z - c[ix.z];
    o.w = l.w - c[ix.w];
    __builtin_nontemporal_store(o, (f32x4*)&out[i4]);
  } else {
    for (int i = i4; i < n; ++i) out[i] = logits[i] - c[index[i]];
  }
}

// ---------------------------------------------------------------------------
extern "C" void kernel_launch(void* const* d_in, const int* in_sizes, int n_in,
                              void* d_out, int out_size, void* d_ws,
                              size_t ws_size, hipStream_t stream) {
  const float* logits = (const float*)d_in[0];
  const int*   index  = (const int*)d_in[1];
  const int*   nseg   = (const int*)d_in[2];  // device scalar (num_segments)
  float*       out    = (float*)d_out;
  const int    n      = in_sizes[0];

  // Partition workspace: [maxEnc (u32 x S)] [sumExp/c (f32 x S)]
  // S is only known on-device, so split ws in half (256B-aligned).
  const size_t half = (ws_size / 2) & ~(size_t)255;
  unsigned* maxEnc = (unsigned*)d_ws;
  float*    sumExp = (float*)((char*)d_ws + half);

  const int tiles  = (n + TILE - 1) / TILE;        // 4096 for n = 2^24
  const int wblks  = (n + BLOCK * 4 - 1) / (BLOCK * 4);
  const int sblks  = 2048;                         // grid-stride over S

  k_init<<<sblks, BLOCK, 0, stream>>>(maxEnc, sumExp, nseg);
  k_segmax<<<tiles, BLOCK, 0, stream>>>(logits, index, maxEnc, n);
  k_sumexp<<<tiles, BLOCK, 0, stream>>>(logits, index, maxEnc, sumExp, n);
  k_segfinal<<<sblks, BLOCK, 0, stream>>>(maxEnc, sumExp, nseg);
  k_write<<<wblks, BLOCK, 0, stream>>>(logits, index, sumExp, out, n);
}